// MemoryUnit_58858231824620
// MI455X (gfx1250) — compile-verified
//
#include <hip/hip_runtime.h>
#include <hip/hip_bf16.h>

// MemoryUnit fused kernel for MI455X (gfx1250, wave32, WMMA).
//
// Roofline: ~1 GB mandatory HBM traffic (features 256MB read, alpha 512MB +
// out 256MB write) -> ~45us floor @ 23.3 TB/s. Logits GEMM is only 137 GFLOP,
// far below even the FP32 WMMA ceiling, so we use V_WMMA_F32_16X16X4_F32 for
// exact f32 math. wn (2MB) and v (2MB) live in the 192MB L2; 64-row blocking
// bounds wn L2 re-reads to ~4GB total.
//
// K is processed in 4 chunks of 128 so the A-strip fragment buffer is only
// 64 VGPRs; the 16x16 f32 accumulators persist in the LDS logits strip
// (per-wave exclusive region, same-wave DS ops are in-order -> no barriers).

typedef float v2f __attribute__((ext_vector_type(2)));
typedef float v4f __attribute__((ext_vector_type(4)));
typedef float v8f __attribute__((ext_vector_type(8)));

#define DDIM 512
#define MDIM 1024
#define ROWS 64          // feature rows per block
#define LROW 1028        // padded floats per logits row (bank-shift half-waves)
#define CAP  64          // max recorded kept entries per row
#define KCHUNK 32        // WMMA K-steps per chunk (128 floats of K)
#define NCHUNK 4         // 4 * 128 = 512 = D
#define L2NORM_EPS 1e-12f

__device__ __forceinline__ v8f wmma_f32_16x16x4(v2f a, v2f b, v8f c) {
  // 8-arg form: (neg_a, A, neg_b, B, c_mod, C, reuse_a, reuse_b)
  return __builtin_amdgcn_wmma_f32_16x16x4_f32(false, a, false, b, (short)0, c,
                                               false, false);
}

// ---------------------------------------------------------------------------
// Kernel 1: wn = l2norm(weight, axis=-1).  One wave per row (M=1024 rows).
// ---------------------------------------------------------------------------
__global__ __launch_bounds__(256) void k_wnorm(const float* __restrict__ w,
                                               float* __restrict__ wn) {
  const int lane = threadIdx.x & 31;
  const int row  = blockIdx.x * 8 + (threadIdx.x >> 5);
  const v4f* wr = (const v4f*)(w + (size_t)row * DDIM);
  v4f vv[4];
  float ssq = 0.f;
#pragma unroll
  for (int i = 0; i < 4; ++i) {
    v4f t = wr[lane + 32 * i];
    vv[i] = t;
    ssq += t.x * t.x + t.y * t.y + t.z * t.z + t.w * t.w;
  }
#pragma unroll
  for (int off = 16; off >= 1; off >>= 1) ssq += __shfl_xor(ssq, off, 32);
  const float sc = 1.0f / fmaxf(sqrtf(ssq), L2NORM_EPS);
  v4f* o = (v4f*)(wn + (size_t)row * DDIM);
#pragma unroll
  for (int i = 0; i < 4; ++i) o[lane + 32 * i] = vv[i] * sc;
}

// ---------------------------------------------------------------------------
// Kernel 2: v = weight @ v_weight.T + v_bias, via f32 WMMA. One wave per
// 16x16 output tile; 64*32 = 2048 tiles.
// ---------------------------------------------------------------------------
__global__ __launch_bounds__(256) void k_vproj(const float* __restrict__ w,
                                               const float* __restrict__ vw,
                                               const float* __restrict__ vb,
                                               float* __restrict__ v) {
  const int lane = threadIdx.x & 31;
  const int tile = blockIdx.x * 8 + (threadIdx.x >> 5);
  const int mt = tile >> 5;   // 0..63  (M/16)
  const int dt = tile & 31;   // 0..31  (D/16)
  const int koff = (lane >> 4) << 1;  // 0 or 2
  const float* ap = w  + (size_t)(mt * 16 + (lane & 15)) * DDIM + koff;
  const float* bp = vw + (size_t)(dt * 16 + (lane & 15)) * DDIM + koff;
  v8f c = {0.f, 0.f, 0.f, 0.f, 0.f, 0.f, 0.f, 0.f};
#pragma unroll 8
  for (int k = 0; k < 128; ++k) {
    v2f a = *(const v2f*)(ap + 4 * k);
    v2f b = *(const v2f*)(bp + 4 * k);
    c = wmma_f32_16x16x4(a, b, c);
  }
  const int col = dt * 16 + (lane & 15);
  const float bias = vb[col];
  const int rbase = mt * 16 + 8 * (lane >> 4);
#pragma unroll
  for (int j = 0; j < 8; ++j)
    v[(size_t)(rbase + j) * DDIM + col] = c[j] + bias;
}

// ---------------------------------------------------------------------------
// Kernel 3: fused logits GEMM + softmax/top8/renorm + sparse out.
// Block = 256 threads (8 waves), owns 64 feature rows (4 strips of 16).
// Wave w -> strip w>>1, M-half w&1 (32 of 64 M-tiles).
// ---------------------------------------------------------------------------
__global__ __launch_bounds__(256) void k_main(const float* __restrict__ features,
                                              const float* __restrict__ wn,
                                              const float* __restrict__ v,
                                              float* __restrict__ out,
                                              float* __restrict__ alpha) {
  extern __shared__ float smem[];
  float* logits  = smem;                                // ROWS * LROW floats
  int*   keptIdx = (int*)(smem + ROWS * LROW);          // 8 * CAP
  float* keptE   = (float*)(keptIdx + 8 * CAP);         // 8 * CAP

  const int lane  = threadIdx.x & 31;
  const int wave  = threadIdx.x >> 5;
  const int strip = wave >> 1;
  const int mhalf = wave & 1;
  const int r0    = blockIdx.x * ROWS;

  const int koff = (lane >> 4) << 1;  // 0 or 2
  const float* ap_base =
      features + (size_t)(r0 + strip * 16 + (lane & 15)) * DDIM + koff;
  const int rbase = strip * 16 + 8 * (lane >> 4);

  float ssq = 0.f;    // running sum of squares of this lane's A elements
  float invf = 0.f;   // 1/||feature row||, valid in the last chunk

#pragma unroll 1
  for (int kc = 0; kc < NCHUNK; ++kc) {
    // ---- Load A chunk (16 rows x 128 K) into WMMA fragments: 64 VGPRs.
    v2f areg[KCHUNK];
#pragma unroll
    for (int k = 0; k < KCHUNK; ++k) {
      v2f a = *(const v2f*)(ap_base + 4 * (kc * KCHUNK + k));
      areg[k] = a;
      ssq += a.x * a.x + a.y * a.y;
    }
    if (kc == NCHUNK - 1) {
      // all of this row's elements seen (lane r + lane r+16 hold the row)
      const float tot = ssq + __shfl_xor(ssq, 16, 32);
      invf = 1.0f / fmaxf(sqrtf(tot), L2NORM_EPS);
    }

    // ---- 32 M-tiles; C accumulates in the LDS logits strip.
#pragma unroll 1
    for (int mi = 0; mi < 32; ++mi) {
      const int mt = mhalf * 32 + mi;
      const float* bp =
          wn + (size_t)(mt * 16 + (lane & 15)) * DDIM + koff + 4 * kc * KCHUNK;
      if (mi < 31)  // prefetch next tile's wn rows (gfx1250 global_prefetch)
        __builtin_prefetch(bp + 16 * DDIM, 0, 0);

      const int colbase = mt * 16 + (lane & 15);
      v8f c;
      if (kc == 0) {
        c = (v8f){0.f, 0.f, 0.f, 0.f, 0.f, 0.f, 0.f, 0.f};
      } else {
#pragma unroll
        for (int j = 0; j < 8; ++j)
          c[j] = logits[(rbase + j) * LROW + colbase];
      }
#pragma unroll
      for (int k = 0; k < KCHUNK; ++k) {
        v2f b = *(const v2f*)(bp + 4 * k);
        c = wmma_f32_16x16x4(areg[k], b, c);
      }
      if (kc == NCHUNK - 1) {
#pragma unroll
        for (int j = 0; j < 8; ++j) {
          const float sc = __shfl(invf, j + ((lane >> 4) << 3), 32);
          logits[(rbase + j) * LROW + colbase] = c[j] * sc;
        }
      } else {
#pragma unroll
        for (int j = 0; j < 8; ++j)
          logits[(rbase + j) * LROW + colbase] = c[j];
      }
    }
  }
  __syncthreads();

  // ---- Phase 2: per-row top8 / exp / L2-renorm / alpha + sparse out.
  for (int rr = 0; rr < 8; ++rr) {
    const int r = wave * 8 + rr;
    float* Lrow = logits + (size_t)r * LROW;

    // per-lane sorted (desc) top-8 over 32 strided columns
    float t[8];
#pragma unroll
    for (int j = 0; j < 8; ++j) t[j] = -3.402823e38f;
#pragma unroll
    for (int i = 0; i < 32; ++i) {
      float a = Lrow[lane + 32 * i];
#pragma unroll
      for (int j = 0; j < 8; ++j) {
        const float mx = fmaxf(t[j], a);
        a = fminf(t[j], a);
        t[j] = mx;
      }
    }
    // 5 xor-merges: exact top-8 of 1024 (duplicates handled correctly)
#pragma unroll
    for (int s = 0; s < 5; ++s) {
      const int off = 16 >> s;
      float h[8];
#pragma unroll
      for (int j = 0; j < 8; ++j) {
        const float p = __shfl_xor(t[7 - j], off, 32);
        h[j] = fmaxf(t[j], p);  // bitonic top-half of the 16-way merge
      }
#define CSWAP(x, y)                       \
  { const float mx = fmaxf(h[x], h[y]);   \
    h[y] = fminf(h[x], h[y]); h[x] = mx; }
      CSWAP(0, 4) CSWAP(1, 5) CSWAP(2, 6) CSWAP(3, 7)
      CSWAP(0, 2) CSWAP(1, 3) CSWAP(4, 6) CSWAP(5, 7)
      CSWAP(0, 1) CSWAP(2, 3) CSWAP(4, 5) CSWAP(6, 7)
#undef CSWAP
#pragma unroll
      for (int j = 0; j < 8; ++j) t[j] = h[j];
    }
    const float thr  = t[7];
    const float rmax = t[0];

    // threshold + exp pass (softmax denominator cancels under L2 renorm)
    int cnt = 0;
    float ssq2 = 0.f;
#pragma unroll 8
    for (int i = 0; i < 32; ++i) {
      const int col = lane + 32 * i;
      const float x = Lrow[col];
      const bool keep = (x >= thr);
      const float e = keep ? __expf(x - rmax) : 0.0f;
      Lrow[col] = e;
      ssq2 += e * e;
      const unsigned m32 = (unsigned)__ballot(keep);
      const int pos = cnt + __popc(m32 & ((1u << lane) - 1u));
      if (keep && pos < CAP) {
        keptIdx[wave * CAP + pos] = col;
        keptE[wave * CAP + pos]   = e;
      }
      cnt += __popc(m32);
    }
#pragma unroll
    for (int off = 16; off >= 1; off >>= 1) ssq2 += __shfl_xor(ssq2, off, 32);
    const float rn = 1.0f / fmaxf(sqrtf(ssq2), L2NORM_EPS);

    // dense alpha row write (coalesced float4)
    float* ag = alpha + (size_t)(r0 + r) * MDIM;
#pragma unroll
    for (int i = 0; i < 8; ++i) {
      const int q = lane + 32 * i;
      v4f e4 = *(const v4f*)(Lrow + 4 * q);
      *(v4f*)(ag + 4 * q) = e4 * rn;
    }

    // sparse out row: out[r] = sum_j alpha_j * v[j]
    const int kn = cnt < CAP ? cnt : CAP;
    v4f acc[4];
    const v4f z = {0.f, 0.f, 0.f, 0.f};
#pragma unroll
    for (int i = 0; i < 4; ++i) acc[i] = z;
    for (int k2 = 0; k2 < kn; ++k2) {
      const float ev = keptE[wave * CAP + k2] * rn;   // wave-uniform LDS read
      const int idx  = keptIdx[wave * CAP + k2];
      const v4f* vrow = (const v4f*)(v + (size_t)idx * DDIM);
#pragma unroll
      for (int i = 0; i < 4; ++i) acc[i] += ev * vrow[lane + 32 * i];
    }
    float* og = out + (size_t)(r0 + r) * DDIM;
#pragma unroll
    for (int i = 0; i < 4; ++i) *(v4f*)(og + 4 * (lane + 32 * i)) = acc[i];
  }
}

// ---------------------------------------------------------------------------
extern "C" void kernel_launch(void* const* d_in, const int* in_sizes, int n_in,
                              void* d_out, int out_size, void* d_ws, size_t ws_size,
                              hipStream_t stream) {
  const float* features = (const float*)d_in[0];
  const float* weight   = (const float*)d_in[1];
  const float* v_weight = (const float*)d_in[2];
  const float* v_bias   = (const float*)d_in[3];
  const int N = in_sizes[0] / DDIM;

  float* wn = (float*)d_ws;                 // M*D floats (2 MB)
  float* v  = wn + (size_t)MDIM * DDIM;     // M*D floats (2 MB)

  float* out   = (float*)d_out;             // N*D
  float* alpha = out + (size_t)N * DDIM;    // N*M

  k_wnorm<<<MDIM / 8, 256, 0, stream>>>(weight, wn);
  k_vproj<<<(MDIM / 16) * (DDIM / 16) / 8, 256, 0, stream>>>(weight, v_weight,
                                                             v_bias, v);
  const size_t smem = (size_t)ROWS * LROW * sizeof(float) + 8 * CAP * 8;
  k_main<<<N / ROWS, 256, smem, stream>>>(features, wn, v, out, alpha);
}